// sub_model_59433757442324
// MI455X (gfx1250) — compile-verified
//
#include <hip/hip_runtime.h>

// SS2D (VMamba selective scan 2D) for gfx1250.
//   1) proj_wmma:    x_dbl = W(38x192) @ xs(192x4096) per (b,k) via V_WMMA_F32_16X16X4_F32
//                    (3 M-tiles per wave -> B-fragment reuse, branch-free A loads)
//   2) scan_phase1:  per-chunk (prod dA, affine Q); x_dbl tiles staged via TENSOR_LOAD_TO_LDS
//   3) chunk_prefix: sequential combine of 32 chunk summaries
//   4) scan_phase3:  rescan chunks from entry state, emit ys (+ D*u); TDM staging
//   5) merge:        CrossMerge

typedef float v2f __attribute__((ext_vector_type(2)));
typedef float v8f __attribute__((ext_vector_type(8)));
typedef unsigned u32x4 __attribute__((ext_vector_type(4)));
typedef int i32x4 __attribute__((ext_vector_type(4)));
typedef int i32x8 __attribute__((ext_vector_type(8)));

constexpr int Bz  = 4;
constexpr int Dm  = 192;
constexpr int Hh  = 64;
constexpr int Ww  = 64;
constexpr int Lsq = Hh * Ww;        // 4096
constexpr int Ksc = 4;              // directions
constexpr int Nst = 16;             // d_state
constexpr int Rdt = 6;              // dt_rank
constexpr int Cc  = Rdt + 2 * Nst;  // 38 rows of x_dbl
constexpr int Sch = 32;             // scan chunks
constexpr int CL  = Lsq / Sch;      // 128 steps per chunk
constexpr int Tt  = 64;             // LDS staging tile (steps)

__device__ __forceinline__ int dir_map(int k, int l) {
  int ll = (k & 2) ? (Lsq - 1 - l) : l;          // k=2,3: reversed
  if (k & 1) ll = ((ll & 63) << 6) | (ll >> 6);  // k=1,3: 64x64 transpose
  return ll;
}

// ---------------------------------------------------------------------------
// TDM: stage a Cc x Tt tile of x_dbl (row stride Lsq floats) into LDS.
// D# bitfields per CDNA5 ISA ch.8 (group0 128b, group1 256b); 2D tile so
// groups 2/3 are zero. data_size=4B, tile row-major -> LDS packed Cc*Tt.
// clang-23 toolchain: 6-arg builtin (g0, g1, g2, g3, g4, cpol).
// ---------------------------------------------------------------------------
__device__ __forceinline__ void tdm_load_tile(const float* gsrc, void* ldsdst) {
  const unsigned long long ga = (unsigned long long)(const void*)gsrc;
  const unsigned lo = (unsigned)(unsigned long long)ldsdst;  // LDS byte offset
  u32x4 g0;
  g0[0] = 1u;                                   // count=1 (valid), user mode
  g0[1] = lo;                                   // lds_addr  (bits 63:32)
  g0[2] = (unsigned)(ga & 0xffffffffu);         // global_addr[31:0]
  g0[3] = (unsigned)((ga >> 32) & 0x01ffffffu)  // global_addr[56:32]
        | (2u << 30);                           // type=2 ("image")
  i32x8 g1;
  g1[0] = (2 << 16);      // data_size=2 -> 4 bytes; wg_mask=0 (not in cluster)
  g1[1] = (Tt << 16);     // tensor_dim0[15:0]  (bits 63:48)
  g1[2] = (Cc << 16);     // tensor_dim1[15:0]  (bits 111:80 -> dw2[31:16])
  g1[3] = (Tt << 16);     // tile_dim0          (bits 127:112)
  g1[4] = Cc;             // tile_dim1          (bits 143:128); tile_dim2=0
  g1[5] = Lsq;            // tensor_dim0_stride[31:0] (bits 191:160)
  g1[6] = 0;              // stride hi / tensor_dim1_stride (unused, 2D)
  g1[7] = 0;
  const i32x4 gz4 = {};
  const i32x8 gz8 = {};
  __builtin_amdgcn_tensor_load_to_lds(g0, g1, gz4, gz4, gz8, 0);
}

// ---------------------------------------------------------------------------
// 1) Projection GEMM. One wave: 3 M-tiles (48 rows, 38 valid) x 16 L-cols,
//    K-loop over 192 d in steps of 4. B fragment shared by the 3 WMMAs.
// ---------------------------------------------------------------------------
__global__ void proj_wmma_kernel(const float* __restrict__ x,
                                 const float* __restrict__ W,
                                 float* __restrict__ xdbl) {
  const int lane   = threadIdx.x & 31;
  const int laneLo = lane & 15;
  const int laneHi = lane >> 4;
  const int wid = blockIdx.x * (blockDim.x >> 5) + (threadIdx.x >> 5);
  const int lt  = wid & 255;          // 256 L-tiles
  const int bk  = wid >> 8;           // b*K + k
  const int b   = bk >> 2;
  const int k   = bk & 3;

  const float* Wk = W + (size_t)k * Cc * Dm;
  const float* Wrow0 = Wk + (size_t)(laneLo) * Dm;        // rows  0..15
  const float* Wrow1 = Wk + (size_t)(16 + laneLo) * Dm;   // rows 16..31
  const int m2 = 32 + laneLo;                             // rows 32..47 (pad)
  const float scale2 = (m2 < Cc) ? 1.f : 0.f;
  const float* Wrow2 = Wk + (size_t)(m2 < Cc ? m2 : 0) * Dm;

  const int l  = lt * 16 + laneLo;
  const int sp = dir_map(k, l);
  const float* xb = x + (size_t)b * Dm * Lsq + sp;

  v8f acc0 = {}, acc1 = {}, acc2 = {};
  for (int d0 = 0; d0 < Dm; d0 += 4) {
    const int dd = d0 + laneHi * 2;   // this lane's K-indices (dd, dd+1)
    v2f bb;
    bb.x = xb[(size_t)dd * Lsq];
    bb.y = xb[(size_t)(dd + 1) * Lsq];
    v2f a0 = *(const v2f*)(Wrow0 + dd);
    v2f a1 = *(const v2f*)(Wrow1 + dd);
    v2f a2 = *(const v2f*)(Wrow2 + dd);
    a2 *= scale2;
    acc0 = __builtin_amdgcn_wmma_f32_16x16x4_f32(false, a0, false, bb,
                                                 (short)0, acc0, false, false);
    acc1 = __builtin_amdgcn_wmma_f32_16x16x4_f32(false, a1, false, bb,
                                                 (short)0, acc1, false, false);
    acc2 = __builtin_amdgcn_wmma_f32_16x16x4_f32(false, a2, false, bb,
                                                 (short)0, acc2, false, false);
  }

  float* outb = xdbl + (size_t)bk * Cc * Lsq + (size_t)lt * 16 + laneLo;
#pragma unroll
  for (int i = 0; i < 8; ++i) {                 // tiles 0,1: rows always valid
    const int mo0 = i + laneHi * 8;
    outb[(size_t)mo0 * Lsq] = acc0[i];
    outb[(size_t)(16 + mo0) * Lsq] = acc1[i];
  }
#pragma unroll
  for (int i = 0; i < 8; ++i) {                 // tile 2: rows 32..47, C=38
    const int mo = 32 + i + laneHi * 8;
    if (mo < Cc) outb[(size_t)mo * Lsq] = acc2[i];
  }
}

__device__ __forceinline__ float softplus_f(float v) {
  return (v > 20.f) ? v : log1pf(__expf(v));
}

// ---------------------------------------------------------------------------
// 2) Phase 1: per chunk, P[n] = prod(dA), Q[n] = h after chunk with h0=0.
//    block = (b,k,chunk); thread = channel d (192 threads). TDM staging.
// ---------------------------------------------------------------------------
__global__ void scan_phase1_kernel(const float* __restrict__ x,
                                   const float* __restrict__ xdbl,
                                   const float* __restrict__ dtw,
                                   const float* __restrict__ dtb,
                                   const float* __restrict__ Alog,
                                   float* __restrict__ Pc,
                                   float* __restrict__ Qc) {
  __shared__ float lds[Cc * Tt];
  const int d  = threadIdx.x;           // 0..191
  const int s  = blockIdx.x % Sch;
  const int bk = blockIdx.x / Sch;
  const int b  = bk >> 2;
  const int k  = bk & 3;

  float wr[Rdt];
#pragma unroll
  for (int r = 0; r < Rdt; ++r) wr[r] = dtw[((size_t)k * Dm + d) * Rdt + r];
  const float bias = dtb[k * Dm + d];
  float Areg[Nst];
#pragma unroll
  for (int n = 0; n < Nst; ++n)
    Areg[n] = -__expf(Alog[((size_t)(k * Dm + d)) * Nst + n]);

  float h[Nst], P[Nst];
#pragma unroll
  for (int n = 0; n < Nst; ++n) { h[n] = 0.f; P[n] = 1.f; }

  const float* xdb = xdbl + (size_t)bk * Cc * Lsq;
  const float* xch = x + (size_t)b * Dm * Lsq + (size_t)d * Lsq;

  for (int t0 = 0; t0 < CL; t0 += Tt) {
    const int l0 = s * CL + t0;
    __syncthreads();                          // previous tile fully consumed
    if (threadIdx.x < 32) {                   // wave 0 drives the TDM
      tdm_load_tile(xdb + l0, lds);
      __builtin_amdgcn_s_wait_tensorcnt(0);
    }
    __syncthreads();
    for (int j = 0; j < Tt; ++j) {
      const int l = l0 + j;
      float dv = bias;
#pragma unroll
      for (int r = 0; r < Rdt; ++r) dv += wr[r] * lds[r * Tt + j];
      const float delta = softplus_f(dv);
      const float u  = xch[dir_map(k, l)];
      const float du = delta * u;
#pragma unroll
      for (int n = 0; n < Nst; ++n) {
        const float dA = __expf(delta * Areg[n]);
        const float Bn = lds[(Rdt + n) * Tt + j];
        h[n] = dA * h[n] + du * Bn;
        P[n] *= dA;
      }
    }
  }
  float* Pp = Pc + ((size_t)bk * Dm + d) * Nst * Sch + s;
  float* Qp = Qc + ((size_t)bk * Dm + d) * Nst * Sch + s;
#pragma unroll
  for (int n = 0; n < Nst; ++n) {
    Pp[(size_t)n * Sch] = P[n];
    Qp[(size_t)n * Sch] = h[n];
  }
}

// ---------------------------------------------------------------------------
// 3) Sequential combine over 32 chunk summaries: entry state per chunk.
// ---------------------------------------------------------------------------
__global__ void chunk_prefix_kernel(const float* __restrict__ Pc,
                                    const float* __restrict__ Qc,
                                    float* __restrict__ h0c) {
  const int gid = blockIdx.x * blockDim.x + threadIdx.x;
  if (gid >= Bz * Ksc * Dm * Nst) return;
  const float* Pp = Pc + (size_t)gid * Sch;
  const float* Qp = Qc + (size_t)gid * Sch;
  float* hp = h0c + (size_t)gid * Sch;
  float h = 0.f;
  for (int s = 0; s < Sch; ++s) {
    hp[s] = h;
    h = Pp[s] * h + Qp[s];
  }
}

// ---------------------------------------------------------------------------
// 4) Phase 3: rescan each chunk from its entry state, emit ys = y + D*u.
// ---------------------------------------------------------------------------
__global__ void scan_phase3_kernel(const float* __restrict__ x,
                                   const float* __restrict__ xdbl,
                                   const float* __restrict__ dtw,
                                   const float* __restrict__ dtb,
                                   const float* __restrict__ Alog,
                                   const float* __restrict__ Dsv,
                                   const float* __restrict__ h0c,
                                   float* __restrict__ ys) {
  __shared__ float lds[Cc * Tt];
  const int d  = threadIdx.x;
  const int s  = blockIdx.x % Sch;
  const int bk = blockIdx.x / Sch;
  const int b  = bk >> 2;
  const int k  = bk & 3;

  float wr[Rdt];
#pragma unroll
  for (int r = 0; r < Rdt; ++r) wr[r] = dtw[((size_t)k * Dm + d) * Rdt + r];
  const float bias = dtb[k * Dm + d];
  const float Dval = Dsv[k * Dm + d];
  float Areg[Nst];
#pragma unroll
  for (int n = 0; n < Nst; ++n)
    Areg[n] = -__expf(Alog[((size_t)(k * Dm + d)) * Nst + n]);

  float h[Nst];
#pragma unroll
  for (int n = 0; n < Nst; ++n)
    h[n] = h0c[(((size_t)bk * Dm + d) * Nst + n) * Sch + s];

  const float* xdb = xdbl + (size_t)bk * Cc * Lsq;
  const float* xch = x + (size_t)b * Dm * Lsq + (size_t)d * Lsq;
  float* yout = ys + ((size_t)bk * Dm + d) * Lsq;

  for (int t0 = 0; t0 < CL; t0 += Tt) {
    const int l0 = s * CL + t0;
    __syncthreads();
    if (threadIdx.x < 32) {
      tdm_load_tile(xdb + l0, lds);
      __builtin_amdgcn_s_wait_tensorcnt(0);
    }
    __syncthreads();
    for (int j = 0; j < Tt; ++j) {
      const int l = l0 + j;
      float dv = bias;
#pragma unroll
      for (int r = 0; r < Rdt; ++r) dv += wr[r] * lds[r * Tt + j];
      const float delta = softplus_f(dv);
      const float u  = xch[dir_map(k, l)];
      const float du = delta * u;
      float y = 0.f;
#pragma unroll
      for (int n = 0; n < Nst; ++n) {
        const float dA = __expf(delta * Areg[n]);
        const float Bn = lds[(Rdt + n) * Tt + j];
        const float Cn = lds[(Rdt + Nst + n) * Tt + j];
        h[n] = dA * h[n] + du * Bn;
        y += h[n] * Cn;
      }
      yout[l] = y + Dval * u;
    }
  }
}

// ---------------------------------------------------------------------------
// 5) CrossMerge
// ---------------------------------------------------------------------------
__global__ void merge_kernel(const float* __restrict__ ys,
                             float* __restrict__ out) {
  const int gid = blockIdx.x * blockDim.x + threadIdx.x;
  if (gid >= Bz * Dm * Lsq) return;
  const int l  = gid % Lsq;
  const int bd = gid / Lsq;            // b*Dm + d
  const int b  = bd / Dm;
  const int d  = bd % Dm;
  const int lT = ((l & 63) << 6) | (l >> 6);
  const size_t strideK = (size_t)Dm * Lsq;
  const size_t base = (size_t)(b * Ksc) * strideK + (size_t)d * Lsq;
  const float v = ys[base + 0 * strideK + l]
                + ys[base + 2 * strideK + (Lsq - 1 - l)]
                + ys[base + 1 * strideK + lT]
                + ys[base + 3 * strideK + (Lsq - 1 - lT)];
  out[gid] = v;
}

// ---------------------------------------------------------------------------
extern "C" void kernel_launch(void* const* d_in, const int* in_sizes, int n_in,
                              void* d_out, int out_size, void* d_ws, size_t ws_size,
                              hipStream_t stream) {
  const float* x    = (const float*)d_in[0];   // (B, D, H, W)
  const float* xpw  = (const float*)d_in[1];   // (K, 38, D)
  const float* dtw  = (const float*)d_in[2];   // (K, D, R)
  const float* dtb  = (const float*)d_in[3];   // (K, D)
  const float* Alog = (const float*)d_in[4];   // (K*D, N)
  const float* Dsv  = (const float*)d_in[5];   // (K*D,)
  float* out = (float*)d_out;

  float* ws = (float*)d_ws;
  const size_t XDBL = (size_t)Bz * Ksc * Cc * Lsq;
  const size_t YS   = (size_t)Bz * Ksc * Dm * Lsq;
  const size_t PQ   = (size_t)Bz * Ksc * Dm * Nst * Sch;
  float* xdbl = ws;
  float* ysb  = xdbl + XDBL;
  float* Pc   = ysb + YS;
  float* Qc   = Pc + PQ;
  float* h0c  = Qc + PQ;

  // 1) projections: 16 bk * 256 ltiles waves (3 M-tiles per wave), 4 waves/block
  proj_wmma_kernel<<<(16 * 256) / 4, 128, 0, stream>>>(x, xpw, xdbl);
  // 2) chunked scan phase 1
  scan_phase1_kernel<<<Bz * Ksc * Sch, Dm, 0, stream>>>(x, xdbl, dtw, dtb, Alog, Pc, Qc);
  // 3) chunk prefix combine
  chunk_prefix_kernel<<<(Bz * Ksc * Dm * Nst + 255) / 256, 256, 0, stream>>>(Pc, Qc, h0c);
  // 4) chunked scan phase 3 (emit ys)
  scan_phase3_kernel<<<Bz * Ksc * Sch, Dm, 0, stream>>>(x, xdbl, dtw, dtb, Alog, Dsv, h0c, ysb);
  // 5) cross-merge
  merge_kernel<<<(Bz * Dm * Lsq + 255) / 256, 256, 0, stream>>>(ysb, out);
}